// EmbedMatcher_11132555231899
// MI455X (gfx1250) — compile-verified
//
#include <hip/hip_runtime.h>
#include <math.h>

// ---------------- constants from the reference ----------------
#define NUM_ENT 100000
#define NUM_REL 5000
#define D_      100
#define B_      256
#define FEW_    3
#define NN_     200
#define STEPS_  4
#define DM_     200   // 2*D
#define DI_     400   // 2*DM
#define G4_     1600  // 4*DI
#define NT_     4     // 16-wide N tiles per wave in the GEMM

typedef float v2f __attribute__((ext_vector_type(2)));
typedef float v8f __attribute__((ext_vector_type(8)));

__device__ __forceinline__ float sigmf(float x) { return 1.0f / (1.0f + expf(-x)); }

// =====================================================================
// K0: view tables. one block per output row; computes nrm(x@Wi + bi) for i=1..3
// =====================================================================
__global__ void views_kernel(const float* __restrict__ ent, const float* __restrict__ rel,
                             const float* __restrict__ W1, const float* __restrict__ b1,
                             const float* __restrict__ W2, const float* __restrict__ b2,
                             const float* __restrict__ W3, const float* __restrict__ b3,
                             const int* __restrict__ query, const int* __restrict__ support,
                             float* __restrict__ rv, float* __restrict__ ev,
                             float* __restrict__ qlv, float* __restrict__ qrv,
                             float* __restrict__ slv, float* __restrict__ srv)
{
    int t = blockIdx.x;
    const float* src;
    float* dst;
    if (t < 5000)        { src = rel + (size_t)t * D_;                       dst = rv  + (size_t)t * 300; }
    else if (t < 10000)  { int e = t - 5000;  src = ent + (size_t)e * D_;    dst = ev  + (size_t)e * 300; }
    else if (t < 10256)  { int b = t - 10000; src = ent + (size_t)query[2*b]   * D_; dst = qlv + (size_t)b * 300; }
    else if (t < 10512)  { int b = t - 10256; src = ent + (size_t)query[2*b+1] * D_; dst = qrv + (size_t)b * 300; }
    else if (t < 10515)  { int f = t - 10512; src = ent + (size_t)support[2*f]   * D_; dst = slv + (size_t)f * 300; }
    else                 { int f = t - 10515; src = ent + (size_t)support[2*f+1] * D_; dst = srv + (size_t)f * 300; }

    __shared__ float x[D_];
    __shared__ float y[D_];
    __shared__ float invn;
    int tid = threadIdx.x;
    if (tid < D_) x[tid] = src[tid];
    __syncthreads();

    const float* Ws[3] = { W1, W2, W3 };
    const float* bs[3] = { b1, b2, b3 };
    for (int w = 0; w < 3; ++w) {
        if (tid < D_) {
            float acc = bs[w][tid];
            const float* Wp = Ws[w];
            for (int k = 0; k < D_; ++k) acc += x[k] * Wp[k * D_ + tid];
            y[tid] = acc;
        }
        __syncthreads();
        if (tid == 0) {
            float s = 0.f;
            for (int d = 0; d < D_; ++d) s += y[d] * y[d];
            invn = 1.0f / fmaxf(sqrtf(s), 1e-12f);
        }
        __syncthreads();
        if (tid < D_) dst[w * D_ + tid] = y[tid] * invn;
        __syncthreads();
    }
}

// =====================================================================
// K1: neigh. grid = nrows*3 blocks (b,view), 256 threads.
// =====================================================================
__global__ void neigh_kernel(const float* __restrict__ leftv, const float* __restrict__ rightv,
                             const int* __restrict__ c_out, const int* __restrict__ c_in,
                             const float* __restrict__ rv, const float* __restrict__ ev,
                             const float* __restrict__ attW1, const float* __restrict__ attb1,
                             const float* __restrict__ attW2, const float* __restrict__ attb2,
                             float* __restrict__ outbuf, int off)
{
    int blk = blockIdx.x;
    int b = blk / 3, v = blk % 3;
    int t = threadIdx.x;

    __shared__ float rr[D_];
    __shared__ float lin[NN_], lout[NN_];
    __shared__ int   iin[NN_], iout[NN_];
    __shared__ float red[256];
    __shared__ float stats[2];

    if (t < D_) rr[t] = rightv[b * 300 + v * D_ + t] - leftv[b * 300 + v * D_ + t];
    __syncthreads();

    float w1r = 0.f, w2r = 0.f;
    for (int d = 0; d < D_; ++d) { float r_ = rr[d]; w1r += r_ * attW1[D_ + d]; w2r += r_ * attW2[D_ + d]; }

    if (t < NN_) {
        int ri = c_in[(b * NN_ + t) * 2 + 0]; int ei = c_in[(b * NN_ + t) * 2 + 1];
        ri = min(ri, NUM_REL - 1); ei = min(ei, NUM_REL - 1);
        iin[t] = ei;
        float a = attb1[0] + w1r;
        const float* rp = rv + (size_t)ri * 300 + v * D_;
        for (int d = 0; d < D_; ++d) a += rp[d] * attW1[d];
        lin[t] = a;

        int ro = c_out[(b * NN_ + t) * 2 + 0]; int eo = c_out[(b * NN_ + t) * 2 + 1];
        ro = min(ro, NUM_REL - 1); eo = min(eo, NUM_REL - 1);
        iout[t] = eo;
        float a2 = attb2[0] + w2r;
        const float* rp2 = rv + (size_t)ro * 300 + v * D_;
        for (int d = 0; d < D_; ++d) a2 += rp2[d] * attW2[d];
        lout[t] = a2;
    }
    __syncthreads();

    // softmax (in)
    red[t] = (t < NN_) ? lin[t] : -1e30f; __syncthreads();
    for (int s = 128; s > 0; s >>= 1) { if (t < s) red[t] = fmaxf(red[t], red[t + s]); __syncthreads(); }
    if (t == 0) stats[0] = red[0];
    __syncthreads();
    float exv = (t < NN_) ? expf(lin[t] - stats[0]) : 0.f;
    if (t < NN_) lin[t] = exv;
    red[t] = exv; __syncthreads();
    for (int s = 128; s > 0; s >>= 1) { if (t < s) red[t] += red[t + s]; __syncthreads(); }
    if (t == 0) stats[1] = red[0];
    __syncthreads();
    if (t < NN_) lin[t] = lin[t] / stats[1];
    __syncthreads();

    // softmax (out)
    red[t] = (t < NN_) ? lout[t] : -1e30f; __syncthreads();
    for (int s = 128; s > 0; s >>= 1) { if (t < s) red[t] = fmaxf(red[t], red[t + s]); __syncthreads(); }
    if (t == 0) stats[0] = red[0];
    __syncthreads();
    exv = (t < NN_) ? expf(lout[t] - stats[0]) : 0.f;
    if (t < NN_) lout[t] = exv;
    red[t] = exv; __syncthreads();
    for (int s = 128; s > 0; s >>= 1) { if (t < s) red[t] += red[t + s]; __syncthreads(); }
    if (t == 0) stats[1] = red[0];
    __syncthreads();
    if (t < NN_) lout[t] = lout[t] / stats[1];
    __syncthreads();

    if (t < D_) {
        float hin = 0.f, hout = 0.f;
        for (int n = 0; n < NN_; ++n) {
            hin  += lin[n]  * ev[(size_t)iin[n]  * 300 + v * D_ + t];
            hout += lout[n] * ev[(size_t)iout[n] * 300 + v * D_ + t];
        }
        float ein = expf(hin), eout = expf(hout);
        float val = tanhf((hin * ein + hout * eout) / (ein + eout));
        outbuf[(size_t)(b * 3 + v) * DM_ + off + t] = val;
    }
}

// =====================================================================
// K2: fp32 WMMA GEMM, software-pipelined.
// C[M,N] = A[M,K]@B[K,N] (+bias[n]) (+add[m,n]) (opt relu)
// One wave per 16x(16*NT_) strip; edge rows/cols clamped (masked at store).
// The K loop prefetches operands for k+4 into a "next" register set BEFORE
// issuing the WMMAs for k, so the loadcnt wait lands a full iteration away
// from the loads instead of directly in front of each matrix op.
// Requires K multiple of 4 and K >= 4 (K=200/400 here).
// =====================================================================
__global__ void gemm16_wmma(const float* __restrict__ A, int lda,
                            const float* __restrict__ Bm, int ldb,
                            const float* __restrict__ bias,
                            const float* __restrict__ add, int ldadd,
                            float* __restrict__ C, int ldc,
                            int M, int N, int K, int relu)
{
    int tn0  = blockIdx.x * (16 * NT_);
    int tm   = blockIdx.y * 16;
    int lane = threadIdx.x;
    int half = lane >> 4;
    int lid  = lane & 15;

    int ma = min(tm + lid, M - 1);
    const float* Arow = A + (size_t)ma * lda + 2 * half;

    const float* Bc[NT_];
#pragma unroll
    for (int j = 0; j < NT_; ++j) {
        int na = min(tn0 + j * 16 + lid, N - 1);
        Bc[j] = Bm + (size_t)(2 * half) * ldb + na;
    }

    v8f acc[NT_];
#pragma unroll
    for (int j = 0; j < NT_; ++j) acc[j] = (v8f){};

    // -------- pipeline prologue: stage k = 0 --------
    v2f a_cur = *(const v2f*)(Arow);
    v2f b_cur[NT_];
#pragma unroll
    for (int j = 0; j < NT_; ++j) {
        b_cur[j].x = Bc[j][0];
        b_cur[j].y = Bc[j][(size_t)ldb];
    }

    // -------- steady state: prefetch k+4, compute k --------
#pragma unroll 2
    for (int k = 0; k < K - 4; k += 4) {
        v2f a_nxt = *(const v2f*)(Arow + k + 4);
        v2f b_nxt[NT_];
#pragma unroll
        for (int j = 0; j < NT_; ++j) {
            const float* bp = Bc[j] + (size_t)(k + 4) * ldb;
            b_nxt[j].x = bp[0];
            b_nxt[j].y = bp[ldb];
        }
#pragma unroll
        for (int j = 0; j < NT_; ++j)
            acc[j] = __builtin_amdgcn_wmma_f32_16x16x4_f32(false, a_cur, false, b_cur[j],
                                                           (short)0, acc[j], false, false);
        a_cur = a_nxt;
#pragma unroll
        for (int j = 0; j < NT_; ++j) b_cur[j] = b_nxt[j];
    }

    // -------- epilogue: last k-step --------
#pragma unroll
    for (int j = 0; j < NT_; ++j)
        acc[j] = __builtin_amdgcn_wmma_f32_16x16x4_f32(false, a_cur, false, b_cur[j],
                                                       (short)0, acc[j], false, false);

#pragma unroll
    for (int j = 0; j < NT_; ++j) {
        int n = tn0 + j * 16 + lid;
        if (n < N) {
            float bv = bias ? bias[n] : 0.f;
            for (int i = 0; i < 8; ++i) {
                int mr = tm + i + (half ? 8 : 0);
                if (mr < M) {
                    float vv = acc[j][i] + bv;
                    if (add)  vv += add[(size_t)mr * ldadd + n];
                    if (relu) vv = fmaxf(vv, 0.f);
                    C[(size_t)mr * ldc + n] = vv;
                }
            }
        }
    }
}

// =====================================================================
// K3: layernorm over last dim (200), one block per row
// =====================================================================
__global__ void ln_kernel(const float* __restrict__ in, const float* __restrict__ g,
                          const float* __restrict__ bb, float* __restrict__ out)
{
    int row = blockIdx.x, t = threadIdx.x;
    __shared__ float xs[DM_];
    __shared__ float red[256];
    __shared__ float mv[2];
    if (t < DM_) xs[t] = in[(size_t)row * DM_ + t];
    __syncthreads();
    red[t] = (t < DM_) ? xs[t] : 0.f; __syncthreads();
    for (int s = 128; s > 0; s >>= 1) { if (t < s) red[t] += red[t + s]; __syncthreads(); }
    if (t == 0) mv[0] = red[0] / (float)DM_;
    __syncthreads();
    float dv = (t < DM_) ? (xs[t] - mv[0]) : 0.f;
    red[t] = dv * dv; __syncthreads();
    for (int s = 128; s > 0; s >>= 1) { if (t < s) red[t] += red[t + s]; __syncthreads(); }
    if (t == 0) mv[1] = red[0] / (float)DM_;
    __syncthreads();
    if (t < DM_)
        out[(size_t)row * DM_ + t] = g[t] * (xs[t] - mv[0]) * rsqrtf(mv[1] + 1e-5f) + bb[t];
}

// K4: sg = mean over FEW of sn_enc rows (f*3+v)
__global__ void sg_mean_kernel(const float* __restrict__ sn_enc, float* __restrict__ sg)
{
    for (int i = threadIdx.x; i < 3 * DM_; i += blockDim.x) {
        int v = i / DM_, d = i % DM_;
        sg[i] = (sn_enc[(0 * 3 + v) * DM_ + d] +
                 sn_enc[(1 * 3 + v) * DM_ + d] +
                 sn_enc[(2 * 3 + v) * DM_ + d]) * (1.0f / 3.0f);
    }
}

__global__ void addvec_kernel(const float* __restrict__ a, const float* __restrict__ b,
                              float* __restrict__ o, int n)
{
    int i = blockIdx.x * blockDim.x + threadIdx.x;
    if (i < n) o[i] = a[i] + b[i];
}

__global__ void zero_kernel(float* __restrict__ p, int n)
{
    int i = blockIdx.x * blockDim.x + threadIdx.x;
    if (i < n) p[i] = 0.f;
}

// =====================================================================
// K5: LSTM step + support attention. one block per batch row.
// =====================================================================
__global__ void lstm_step_kernel(const float* __restrict__ gates,
                                 const float* __restrict__ qbase, int qstride,
                                 const float* __restrict__ supp,  // (3,200) = sg
                                 float* __restrict__ c, float* __restrict__ hr)
{
    int b = blockIdx.x, t = threadIdx.x;
    __shared__ float hs[DM_];
    __shared__ float red[256];
    __shared__ float wv[3];
    __shared__ float dots[3];
    const float* g = gates + (size_t)b * G4_;

    for (int j = t; j < DI_; j += 256) {
        float ig = g[j], fg = g[DI_ + j], gg = g[2 * DI_ + j], og = g[3 * DI_ + j];
        float c2 = sigmf(fg) * c[(size_t)b * DI_ + j] + sigmf(ig) * tanhf(gg);
        c[(size_t)b * DI_ + j] = c2;
        float h2 = sigmf(og) * tanhf(c2);
        if (j < DM_) hs[j] = qbase[(size_t)b * qstride + j] + h2;
    }
    __syncthreads();

    float p0 = 0.f, p1 = 0.f, p2 = 0.f;
    for (int d = t; d < DM_; d += 256) {
        float hv = hs[d];
        p0 += hv * supp[d]; p1 += hv * supp[DM_ + d]; p2 += hv * supp[2 * DM_ + d];
    }
    float ps[3] = { p0, p1, p2 };
    for (int j = 0; j < 3; ++j) {
        red[t] = ps[j]; __syncthreads();
        for (int s = 128; s > 0; s >>= 1) { if (t < s) red[t] += red[t + s]; __syncthreads(); }
        if (t == 0) dots[j] = red[0];
        __syncthreads();
    }
    if (t == 0) {
        float m = fmaxf(dots[0], fmaxf(dots[1], dots[2]));
        float e0 = expf(dots[0] - m), e1 = expf(dots[1] - m), e2 = expf(dots[2] - m);
        float inv = 1.0f / (e0 + e1 + e2);
        wv[0] = e0 * inv; wv[1] = e1 * inv; wv[2] = e2 * inv;
    }
    __syncthreads();
    for (int d = t; d < DM_; d += 256) {
        hr[(size_t)b * DI_ + d]       = hs[d];
        hr[(size_t)b * DI_ + DM_ + d] = wv[0] * supp[d] + wv[1] * supp[DM_ + d] + wv[2] * supp[2 * DM_ + d];
    }
}

// K6: s[b] = dot(hr[b,0:200], sgv)
__global__ void score_kernel(const float* __restrict__ hr, const float* __restrict__ sgv,
                             float* __restrict__ sout)
{
    int b = blockIdx.x, t = threadIdx.x;
    __shared__ float red[256];
    float p = 0.f;
    for (int d = t; d < DM_; d += 256) p += hr[(size_t)b * DI_ + d] * sgv[d];
    red[t] = p; __syncthreads();
    for (int s = 128; s > 0; s >>= 1) { if (t < s) red[t] += red[t + s]; __syncthreads(); }
    if (t == 0) sout[b] = red[0];
}

// K7: out[b] = max over views
__global__ void max3_kernel(const float* __restrict__ s, float* __restrict__ out)
{
    int b = threadIdx.x;
    if (b < B_) out[b] = fmaxf(fmaxf(s[b], s[B_ + b]), s[2 * B_ + b]);
}

// =====================================================================
extern "C" void kernel_launch(void* const* d_in, const int* in_sizes, int n_in,
                              void* d_out, int out_size, void* d_ws, size_t ws_size,
                              hipStream_t stream)
{
    (void)in_sizes; (void)n_in; (void)out_size; (void)ws_size;
    const float* ent_emb = (const float*)d_in[0];
    const float* rel_emb = (const float*)d_in[1];
    const float* W1 = (const float*)d_in[2];  const float* b1 = (const float*)d_in[3];
    const float* W2 = (const float*)d_in[4];  const float* b2 = (const float*)d_in[5];
    const float* W3 = (const float*)d_in[6];  const float* b3 = (const float*)d_in[7];
    const float* attW1 = (const float*)d_in[8];  const float* attb1 = (const float*)d_in[9];
    const float* attW2 = (const float*)d_in[10]; const float* attb2 = (const float*)d_in[11];
    const float* seW1 = (const float*)d_in[12]; const float* seb1 = (const float*)d_in[13];
    const float* seW2 = (const float*)d_in[14]; const float* seb2 = (const float*)d_in[15];
    const float* ln_g = (const float*)d_in[16]; const float* ln_b = (const float*)d_in[17];
    const float* Wih = (const float*)d_in[18];  const float* bih = (const float*)d_in[19];
    const float* Whh = (const float*)d_in[20];  const float* bhh = (const float*)d_in[21];
    const int* query   = (const int*)d_in[22];
    const int* support = (const int*)d_in[23];
    const int* qlc_out = (const int*)d_in[24]; const int* qlc_in = (const int*)d_in[25];
    const int* qrc_out = (const int*)d_in[27]; const int* qrc_in = (const int*)d_in[28];
    const int* slc_out = (const int*)d_in[30]; const int* slc_in = (const int*)d_in[31];
    const int* src_out = (const int*)d_in[33]; const int* src_in = (const int*)d_in[34];
    float* out = (float*)d_out;

    // ---------------- workspace layout (floats) ----------------
    float* ws = (float*)d_ws;
    float* rv_   = ws;  ws += (size_t)NUM_REL * 300;
    float* ev_   = ws;  ws += (size_t)NUM_REL * 300;
    float* qlv   = ws;  ws += (size_t)B_ * 300;
    float* qrv   = ws;  ws += (size_t)B_ * 300;
    float* slv   = ws;  ws += (size_t)FEW_ * 300;
    float* srv   = ws;  ws += (size_t)FEW_ * 300;
    float* qn    = ws;  ws += (size_t)B_ * 3 * DM_;
    float* sn    = ws;  ws += (size_t)FEW_ * 3 * DM_;
    float* h1q   = ws;  ws += (size_t)B_ * 3 * DI_;
    float* hpreq = ws;  ws += (size_t)B_ * 3 * DM_;
    float* qg    = ws;  ws += (size_t)B_ * 3 * DM_;
    float* h1s   = ws;  ws += (size_t)FEW_ * 3 * DI_;
    float* hpres = ws;  ws += (size_t)FEW_ * 3 * DM_;
    float* sn_e  = ws;  ws += (size_t)FEW_ * 3 * DM_;
    float* sg    = ws;  ws += (size_t)3 * DM_;
    float* bsum  = ws;  ws += (size_t)G4_;
    float* P     = ws;  ws += (size_t)B_ * G4_;
    float* gates = ws;  ws += (size_t)B_ * G4_;
    float* cst   = ws;  ws += (size_t)B_ * DI_;
    float* hr    = ws;  ws += (size_t)B_ * DI_;
    float* svw   = ws;  ws += (size_t)3 * B_;

    // ---------------- Phase A: view tables ----------------
    views_kernel<<<10518, 128, 0, stream>>>(ent_emb, rel_emb, W1, b1, W2, b2, W3, b3,
                                            query, support, rv_, ev_, qlv, qrv, slv, srv);

    // ---------------- Phase B: neigh x4 ----------------
    neigh_kernel<<<B_ * 3, 256, 0, stream>>>(qlv, qrv, qlc_out, qlc_in, rv_, ev_,
                                             attW1, attb1, attW2, attb2, qn, 0);
    neigh_kernel<<<B_ * 3, 256, 0, stream>>>(qlv, qrv, qrc_out, qrc_in, rv_, ev_,
                                             attW1, attb1, attW2, attb2, qn, D_);
    neigh_kernel<<<FEW_ * 3, 256, 0, stream>>>(slv, srv, slc_out, slc_in, rv_, ev_,
                                               attW1, attb1, attW2, attb2, sn, 0);
    neigh_kernel<<<FEW_ * 3, 256, 0, stream>>>(slv, srv, src_out, src_in, rv_, ev_,
                                               attW1, attb1, attW2, attb2, sn, D_);

    // ---------------- Phase C: senc (WMMA GEMMs) ----------------
    gemm16_wmma<<<dim3(7, 48), 32, 0, stream>>>(qn, DM_, seW1, DI_, seb1, nullptr, 0,
                                                h1q, DI_, 768, DI_, DM_, 1);
    gemm16_wmma<<<dim3(4, 48), 32, 0, stream>>>(h1q, DI_, seW2, DM_, seb2, qn, DM_,
                                                hpreq, DM_, 768, DM_, DI_, 0);
    ln_kernel<<<768, 256, 0, stream>>>(hpreq, ln_g, ln_b, qg);

    gemm16_wmma<<<dim3(7, 1), 32, 0, stream>>>(sn, DM_, seW1, DI_, seb1, nullptr, 0,
                                               h1s, DI_, 9, DI_, DM_, 1);
    gemm16_wmma<<<dim3(4, 1), 32, 0, stream>>>(h1s, DI_, seW2, DM_, seb2, sn, DM_,
                                               hpres, DM_, 9, DM_, DI_, 0);
    ln_kernel<<<9, 256, 0, stream>>>(hpres, ln_g, ln_b, sn_e);
    sg_mean_kernel<<<1, 256, 0, stream>>>(sn_e, sg);

    // ---------------- Phase D: qenc LSTM (WMMA GEMMs), per view ----------------
    addvec_kernel<<<(G4_ + 255) / 256, 256, 0, stream>>>(bih, bhh, bsum, G4_);
    for (int v = 0; v < 3; ++v) {
        gemm16_wmma<<<dim3(25, 16), 32, 0, stream>>>(qg + v * DM_, 3 * DM_, Wih, G4_,
                                                     bsum, nullptr, 0, P, G4_,
                                                     B_, G4_, DM_, 0);
        zero_kernel<<<(B_ * DI_ + 255) / 256, 256, 0, stream>>>(cst, B_ * DI_);
        zero_kernel<<<(B_ * DI_ + 255) / 256, 256, 0, stream>>>(hr, B_ * DI_);
        for (int step = 0; step < STEPS_; ++step) {
            gemm16_wmma<<<dim3(25, 16), 32, 0, stream>>>(hr, DI_, Whh, G4_,
                                                         nullptr, P, G4_, gates, G4_,
                                                         B_, G4_, DI_, 0);
            lstm_step_kernel<<<B_, 256, 0, stream>>>(gates, qg + v * DM_, 3 * DM_, sg, cst, hr);
        }
        score_kernel<<<B_, 256, 0, stream>>>(hr, sg + v * DM_, svw + v * B_);
    }
    max3_kernel<<<1, 256, 0, stream>>>(svw, out);
}